// MicroMacroArchitectureModel_16784732192990
// MI455X (gfx1250) — compile-verified
//
#include <hip/hip_runtime.h>
#include <hip/hip_bf16.h>

typedef __attribute__((ext_vector_type(2))) float v2f;
typedef __attribute__((ext_vector_type(8))) float v8f;

#define N_NODES 50000
#define N_EDGES 600000
#define HID 128
#define N_GRAPHS 512
#define OUT_DIM 10
#define LN_EPS 1e-5f

static __device__ __forceinline__ void atomic_add_f32(float* p, float v) {
    __hip_atomic_fetch_add(p, v, __ATOMIC_RELAXED, __HIP_MEMORY_SCOPE_AGENT);
}

// ---------------- degree / dinv ----------------
__global__ void k_init_deg(float* deg, int n) {
    int i = blockIdx.x * blockDim.x + threadIdx.x;
    if (i < n) deg[i] = 1.0f;  // self loop
}

__global__ void k_count_deg(const int* __restrict__ dst, float* deg, int e) {
    int i = blockIdx.x * blockDim.x + threadIdx.x;
    if (i < e) atomic_add_f32(&deg[dst[i]], 1.0f);
}

__global__ void k_finalize_dinv(float* deg, int n) {
    int i = blockIdx.x * blockDim.x + threadIdx.x;
    if (i < n) deg[i] = rsqrtf(deg[i]);  // deg >= 1 via self loop
}

// ---------------- WMMA GEMM: C[M x 128] = (sum of NIN inputs) @ W[128 x 128] (+ bias)
// One wave computes rows [16*wave, 16*wave+16) x all 128 cols.
// f32 WMMA 16x16x4: exact-precision matrix path on CDNA5.
// NIN templated so the inner loop has no exec-mask branching.
template <int NIN, bool HAS_BIAS>
__global__ void k_gemm_nodes(const float* __restrict__ A1,
                             const float* __restrict__ A2,
                             const float* __restrict__ A3,
                             const float* __restrict__ W,
                             const float* __restrict__ bias,
                             float* __restrict__ C, int M) {
    int wave = (blockIdx.x * blockDim.x + threadIdx.x) >> 5;
    int lane = threadIdx.x & 31;
    int row0 = wave * 16;
    if (row0 >= M) return;
    int lhalf = lane >> 4;  // 0: lanes 0-15, 1: lanes 16-31
    int lm = lane & 15;

    v8f acc[8] = {};
    const float* r1 = A1 + (size_t)(row0 + lm) * HID;
    const float* r2 = (NIN >= 2) ? A2 + (size_t)(row0 + lm) * HID : nullptr;
    const float* r3 = (NIN >= 3) ? A3 + (size_t)(row0 + lm) * HID : nullptr;

    for (int k0 = 0; k0 < HID; k0 += 4) {
        // A 16x4 layout: lane lm holds M=lm, VGPR pair = K = k0 + 2*lhalf + {0,1}
        int ka = k0 + 2 * lhalf;
        float a0 = r1[ka], a1 = r1[ka + 1];
        if (NIN >= 2) { a0 += r2[ka]; a1 += r2[ka + 1]; }
        if (NIN >= 3) { a0 += r3[ka]; a1 += r3[ka + 1]; }
        v2f a; a.x = a0; a.y = a1;
#pragma unroll
        for (int j = 0; j < 8; ++j) {
            // B 4x16 layout: VGPR v holds row K = k0 + 2*lhalf + v, cols striped on lm
            v2f b;
            b.x = W[(size_t)(ka + 0) * HID + j * 16 + lm];
            b.y = W[(size_t)(ka + 1) * HID + j * 16 + lm];
            acc[j] = __builtin_amdgcn_wmma_f32_16x16x4_f32(
                false, a, false, b, (short)0, acc[j], false, false);
        }
    }
    // C/D layout: acc[j][v] = C[row0 + v + 8*lhalf][16*j + lm]
#pragma unroll
    for (int j = 0; j < 8; ++j) {
        float bj = HAS_BIAS ? bias[j * 16 + lm] : 0.0f;
#pragma unroll
        for (int v = 0; v < 8; ++v) {
            C[(size_t)(row0 + v + 8 * lhalf) * HID + j * 16 + lm] = acc[j][v] + bj;
        }
    }
}

// ---------------- self-loop + bias init: out = dinv[n]^2 * h + b ----------------
__global__ void k_selfloop_bias(const float* __restrict__ h,
                                const float* __restrict__ dinv,
                                const float* __restrict__ bias,
                                float* __restrict__ out, int n_elems) {
    int i = blockIdx.x * blockDim.x + threadIdx.x;
    if (i >= n_elems) return;
    int node = i >> 7;
    int c = i & (HID - 1);
    float di = dinv[node];
    out[i] = di * di * h[i] + bias[c];
}

// ---------------- edge scatter: out[dst] += dinv[s]*dinv[d]*h[src] -------------
// One wave per edge: 32 lanes x 4 floats, coalesced 128B atomic bursts.
__global__ void k_edge_scatter(const float* __restrict__ h,
                               const int* __restrict__ src,
                               const int* __restrict__ dst,
                               const float* __restrict__ dinv,
                               float* __restrict__ out, int e) {
    int wave = (blockIdx.x * blockDim.x + threadIdx.x) >> 5;
    int lane = threadIdx.x & 31;
    if (wave >= e) return;
    int s = src[wave], d = dst[wave];
    float coef = dinv[s] * dinv[d];
    const float* hs = h + (size_t)s * HID;
    float* od = out + (size_t)d * HID;
#pragma unroll
    for (int k = 0; k < 4; ++k) {
        atomic_add_f32(&od[lane + 32 * k], coef * hs[lane + 32 * k]);
    }
}

// ---------------- LayerNorm + ReLU, one wave per node ----------------
__global__ void k_ln_relu(const float* __restrict__ in,
                          const float* __restrict__ g,
                          const float* __restrict__ b,
                          float* __restrict__ out, int n) {
    int wave = (blockIdx.x * blockDim.x + threadIdx.x) >> 5;
    int lane = threadIdx.x & 31;
    if (wave >= n) return;
    const float* row = in + (size_t)wave * HID;
    float v[4], sum = 0.0f, sq = 0.0f;
#pragma unroll
    for (int k = 0; k < 4; ++k) {
        v[k] = row[lane + 32 * k];
        sum += v[k];
        sq += v[k] * v[k];
    }
#pragma unroll
    for (int off = 16; off > 0; off >>= 1) {
        sum += __shfl_xor(sum, off, 32);
        sq += __shfl_xor(sq, off, 32);
    }
    float mean = sum * (1.0f / HID);
    float var = sq * (1.0f / HID) - mean * mean;
    float inv = rsqrtf(var + LN_EPS);
    float* orow = out + (size_t)wave * HID;
#pragma unroll
    for (int k = 0; k < 4; ++k) {
        int c = lane + 32 * k;
        float y = (v[k] - mean) * inv * g[c] + b[c];
        orow[c] = fmaxf(y, 0.0f);
    }
}

// ---------------- pooling ----------------
__global__ void k_zero(float* p, int n) {
    int i = blockIdx.x * blockDim.x + threadIdx.x;
    if (i < n) p[i] = 0.0f;
}

__global__ void k_pool_scatter(const float* __restrict__ last,
                               const int* __restrict__ batch,
                               float* __restrict__ pooled, int n_elems) {
    int i = blockIdx.x * blockDim.x + threadIdx.x;
    if (i >= n_elems) return;
    int node = i >> 7;
    int c = i & (HID - 1);
    atomic_add_f32(&pooled[(size_t)batch[node] * HID + c], last[i]);
}

// ---------------- post MLP: [512,128] @ [128,10] + b ----------------
__global__ void k_post_mlp(const float* __restrict__ pooled,
                           const float* __restrict__ Wp,
                           const float* __restrict__ bp,
                           float* __restrict__ out) {
    int i = blockIdx.x * blockDim.x + threadIdx.x;
    if (i >= N_GRAPHS * OUT_DIM) return;
    int gg = i / OUT_DIM, o = i % OUT_DIM;
    float acc = bp[o];
    const float* pr = pooled + (size_t)gg * HID;
    for (int k = 0; k < HID; ++k) acc += pr[k] * Wp[k * OUT_DIM + o];
    out[i] = acc;
}

extern "C" void kernel_launch(void* const* d_in, const int* in_sizes, int n_in,
                              void* d_out, int out_size, void* d_ws, size_t ws_size,
                              hipStream_t stream) {
    const float* x = (const float*)d_in[0];
    const int* edge_index = (const int*)d_in[1];
    const int* batch = (const int*)d_in[2];
    const float* W_pre = (const float*)d_in[3];
    const float* b_pre = (const float*)d_in[4];
    const float* conv_W = (const float*)d_in[5];
    const float* conv_b = (const float*)d_in[6];
    const float* ln_g = (const float*)d_in[7];
    const float* ln_b = (const float*)d_in[8];
    const float* W_post = (const float*)d_in[9];
    const float* b_post = (const float*)d_in[10];

    const int* src = edge_index;             // row 0
    const int* dst = edge_index + N_EDGES;   // row 1

    // workspace carve-up
    char* ws = (char*)d_ws;
    const size_t SLOT = (size_t)N_NODES * HID * sizeof(float);  // 25.6 MB
    float* dinv = (float*)ws;                                    // 200 KB
    float* pooled = (float*)(ws + 262144);                       // 256 KB
    float* S0 = (float*)(ws + 1048576);            // h0, later: last
    float* S1 = (float*)(ws + 1048576 + 1 * SLOT); // gemm output h
    float* S2 = (float*)(ws + 1048576 + 2 * SLOT); // aggregation buffer
    float* S3 = (float*)(ws + 1048576 + 3 * SLOT); // n0
    float* S4 = (float*)(ws + 1048576 + 4 * SLOT); // n1
    float* S5 = (float*)(ws + 1048576 + 5 * SLOT); // s2l

    const int T = 256;
    const int gemm_blocks = ((N_NODES / 16) * 32 + T - 1) / T;   // 16 rows per wave
    const int elem_blocks = (N_NODES * HID + T - 1) / T;
    const int edge_wave_blocks = (N_EDGES * 32 + T - 1) / T;     // 1 wave per edge
    const int node_wave_blocks = (N_NODES * 32 + T - 1) / T;     // 1 wave per node

    // dinv = rsqrt(deg) with self loops
    k_init_deg<<<(N_NODES + T - 1) / T, T, 0, stream>>>(dinv, N_NODES);
    k_count_deg<<<(N_EDGES + T - 1) / T, T, 0, stream>>>(dst, dinv, N_EDGES);
    k_finalize_dinv<<<(N_NODES + T - 1) / T, T, 0, stream>>>(dinv, N_NODES);

    // pre MLP: h0 = x @ W_pre + b_pre
    k_gemm_nodes<1, true><<<gemm_blocks, T, 0, stream>>>(
        x, nullptr, nullptr, W_pre, b_pre, S0, N_NODES);

    // ---- cell 0: input h0 -> n0 (S3)
    // ---- cell 1: input n0 -> n1 (S4)
    // ---- cell 2: input n0+n1 -> s2l (S5)
    // ---- cell 3: input n0+n1+s2l -> last (S0)
    for (int c = 0; c < 4; ++c) {
        const float* Wc = conv_W + (size_t)c * HID * HID;
        if (c == 0) {
            k_gemm_nodes<1, false><<<gemm_blocks, T, 0, stream>>>(
                S0, nullptr, nullptr, Wc, nullptr, S1, N_NODES);
        } else if (c == 1) {
            k_gemm_nodes<1, false><<<gemm_blocks, T, 0, stream>>>(
                S3, nullptr, nullptr, Wc, nullptr, S1, N_NODES);
        } else if (c == 2) {
            k_gemm_nodes<2, false><<<gemm_blocks, T, 0, stream>>>(
                S3, S4, nullptr, Wc, nullptr, S1, N_NODES);
        } else {
            k_gemm_nodes<3, false><<<gemm_blocks, T, 0, stream>>>(
                S3, S4, S5, Wc, nullptr, S1, N_NODES);
        }
        k_selfloop_bias<<<elem_blocks, T, 0, stream>>>(
            S1, dinv, conv_b + (size_t)c * HID, S2, N_NODES * HID);
        k_edge_scatter<<<edge_wave_blocks, T, 0, stream>>>(S1, src, dst, dinv, S2, N_EDGES);
        float* outb = (c == 0) ? S3 : (c == 1) ? S4 : (c == 2) ? S5 : S0;
        k_ln_relu<<<node_wave_blocks, T, 0, stream>>>(
            S2, ln_g + (size_t)c * HID, ln_b + (size_t)c * HID, outb, N_NODES);
    }

    // sum pooling + post MLP
    k_zero<<<(N_GRAPHS * HID + T - 1) / T, T, 0, stream>>>(pooled, N_GRAPHS * HID);
    k_pool_scatter<<<elem_blocks, T, 0, stream>>>(S0, batch, pooled, N_NODES * HID);
    k_post_mlp<<<(N_GRAPHS * OUT_DIM + T - 1) / T, T, 0, stream>>>(
        pooled, W_post, b_post, (float*)d_out);
}